// GraphModel_17119739642111
// MI455X (gfx1250) — compile-verified
//
#include <hip/hip_runtime.h>

typedef __attribute__((ext_vector_type(2))) float v2f;
typedef __attribute__((ext_vector_type(8))) float v8f;

#define ALPHA_C 0.1f
#define ONE_MINUS_ALPHA 0.9f
#define K_STEPS 10

// ---------------------------------------------------------------------------
// Degree / norm precompute
// ---------------------------------------------------------------------------
__global__ __launch_bounds__(256) void init_deg_kernel(float* __restrict__ deg, int n) {
  int i = blockIdx.x * blockDim.x + threadIdx.x;
  if (i < n) deg[i] = 1.0f;  // self-loop
}

__global__ __launch_bounds__(256) void count_deg_kernel(const int* __restrict__ dst,
                                                        float* __restrict__ deg, int e) {
  int i = blockIdx.x * blockDim.x + threadIdx.x;
  if (i < e) atomicAdd(&deg[dst[i]], 1.0f);
}

// nrm[e] = (1-alpha) * dinv[src] * dinv[dst]   ((1-alpha) pre-baked: saves a
// multiply in each of the 10 edge-scatter passes)
__global__ __launch_bounds__(256) void edge_norm_kernel(const int* __restrict__ src,
                                                        const int* __restrict__ dst,
                                                        const float* __restrict__ deg,
                                                        float* __restrict__ nrm, int e) {
  int i = blockIdx.x * blockDim.x + threadIdx.x;
  if (i < e) nrm[i] = ONE_MINUS_ALPHA * rsqrtf(deg[src[i]]) * rsqrtf(deg[dst[i]]);
}

// ---------------------------------------------------------------------------
// MLP via fp32 WMMA: h = relu(emb @ w1^T + b1) @ w2^T + b2   (16 rows / wave)
// Branch-free zero padding of B columns 8..15: clamp the row index and scale
// the loaded value by a 0/1 lane mask (avoids per-step saveexec divergence).
// ---------------------------------------------------------------------------
__global__ __launch_bounds__(256) void mlp_wmma_kernel(const float* __restrict__ emb,
                                                       const float* __restrict__ w1,
                                                       const float* __restrict__ b1,
                                                       const float* __restrict__ w2,
                                                       const float* __restrict__ b2,
                                                       float* __restrict__ hbuf,
                                                       float* __restrict__ x0,
                                                       int n_nodes) {
  __shared__ float lds[8][16][16];  // per-wave X1 staging tile
  const int lane = threadIdx.x & 31;
  const int wv   = threadIdx.x >> 5;
  const int m    = lane & 15;   // A: row M ; B/C/D: column N
  const int hh   = lane >> 4;   // lane half
  const int rbase = blockIdx.x * 128 + wv * 16;

  // ---- Stage 1: X1[16x16] = emb_tile[16x32] @ B1[32x16]  (B cols 8..15 = 0)
  const int   rowm = (rbase + m < n_nodes) ? (rbase + m) : (n_nodes - 1);
  const int   mb   = m & 7;                       // clamped w1 row (always valid)
  const float msk  = (m < 8) ? 1.0f : 0.0f;       // 0/1 column mask, branch-free
  v8f acc = {0.f, 0.f, 0.f, 0.f, 0.f, 0.f, 0.f, 0.f};
#pragma unroll
  for (int ks = 0; ks < 8; ++ks) {
    const int k0 = ks * 4 + 2 * hh;     // A lane holds K = k0, k0+1
    v2f a, b;
    a.x = emb[rowm * 32 + k0];
    a.y = emb[rowm * 32 + k0 + 1];
    b.x = msk * w1[mb * 32 + k0];       // B[k][n] = w1[n][k], zero for n>=8
    b.y = msk * w1[mb * 32 + k0 + 1];
    acc = __builtin_amdgcn_wmma_f32_16x16x4_f32(false, a, false, b, (short)0,
                                                acc, false, false);
  }
  const float bias1 = msk * b1[mb];
#pragma unroll
  for (int r = 0; r < 8; ++r) {
    float t = fmaxf(acc[r] + bias1, 0.0f);   // relu; padded cols stay 0
    lds[wv][r + 8 * hh][m] = t;              // element X1[M = r+8*hh][N = m]
  }
  __syncthreads();

  // ---- Stage 2: X2[16x16] = X1[16x8] @ B2[8x16], B2[k][n] = w2[n][k]
  v8f acc2 = {0.f, 0.f, 0.f, 0.f, 0.f, 0.f, 0.f, 0.f};
#pragma unroll
  for (int ks = 0; ks < 2; ++ks) {
    const int k0 = ks * 4 + 2 * hh;     // K in [0,8)
    v2f a, b;
    a.x = lds[wv][m][k0];
    a.y = lds[wv][m][k0 + 1];
    b.x = w2[m * 8 + k0];
    b.y = w2[m * 8 + k0 + 1];
    acc2 = __builtin_amdgcn_wmma_f32_16x16x4_f32(false, a, false, b, (short)0,
                                                 acc2, false, false);
  }
  const float bias2 = b2[m];
#pragma unroll
  for (int r = 0; r < 8; ++r) {
    const int row = rbase + r + 8 * hh;
    if (row < n_nodes) {
      const float t = acc2[r] + bias2;
      hbuf[row * 16 + m] = t;
      x0[row * 16 + m]   = t;
    }
  }
}

// ---------------------------------------------------------------------------
// APPNP round = node init (fold alpha*h + self-loop) + edge scatter-add
// ---------------------------------------------------------------------------
__global__ __launch_bounds__(256) void node_init_kernel(const float* __restrict__ hbuf,
                                                        const float* __restrict__ xin,
                                                        const float* __restrict__ deg,
                                                        float* __restrict__ xout,
                                                        int n_nodes) {
  int tid = blockIdx.x * blockDim.x + threadIdx.x;   // one float4 quad per thread
  if (tid < n_nodes * 4) {
    const int n = tid >> 2;
    const float coef = ONE_MINUS_ALPHA / deg[n];     // (1-a) * dinv[n]^2 (self loop)
    const float4 hv = ((const float4*)hbuf)[tid];
    const float4 xv = ((const float4*)xin)[tid];
    float4 o;
    o.x = ALPHA_C * hv.x + coef * xv.x;
    o.y = ALPHA_C * hv.y + coef * xv.y;
    o.z = ALPHA_C * hv.z + coef * xv.z;
    o.w = ALPHA_C * hv.w + coef * xv.w;
    ((float4*)xout)[tid] = o;
  }
}

__global__ __launch_bounds__(256) void edge_scatter_kernel(const int* __restrict__ src,
                                                           const int* __restrict__ dst,
                                                           const float* __restrict__ nrm,
                                                           const float* __restrict__ xin,
                                                           float* __restrict__ xout,
                                                           int n_edges) {
  int tid = blockIdx.x * blockDim.x + threadIdx.x;   // 4 threads per edge
  if (tid < n_edges * 4) {
    const int e = tid >> 2;
    const int q = tid & 3;
    const int s = src[e];
    const int d = dst[e];
    const float w = nrm[e];                          // already includes (1-alpha)
    const float4 v = ((const float4*)xin)[s * 4 + q];
    float* outp = xout + (size_t)d * 16 + q * 4;
    atomicAdd(outp + 0, w * v.x);
    atomicAdd(outp + 1, w * v.y);
    atomicAdd(outp + 2, w * v.z);
    atomicAdd(outp + 3, w * v.w);
  }
}

// ---------------------------------------------------------------------------
// Edge dot-product scoring
// ---------------------------------------------------------------------------
__global__ __launch_bounds__(256) void score_kernel(const int* __restrict__ src,
                                                    const int* __restrict__ dst,
                                                    const float* __restrict__ x,
                                                    float* __restrict__ pred,
                                                    int n_edges) {
  int e = blockIdx.x * blockDim.x + threadIdx.x;
  if (e < n_edges) {
    const float4* xs = (const float4*)x + (size_t)src[e] * 4;
    const float4* xd = (const float4*)x + (size_t)dst[e] * 4;
    float acc = 0.0f;
#pragma unroll
    for (int q = 0; q < 4; ++q) {
      const float4 a = xs[q];
      const float4 b = xd[q];
      acc += a.x * b.x + a.y * b.y + a.z * b.z + a.w * b.w;
    }
    pred[e] = acc;
  }
}

// ---------------------------------------------------------------------------
extern "C" void kernel_launch(void* const* d_in, const int* in_sizes, int n_in,
                              void* d_out, int out_size, void* d_ws, size_t ws_size,
                              hipStream_t stream) {
  const int*   edge = (const int*)d_in[0];    // [2, E] int32
  const float* emb  = (const float*)d_in[1];  // [N, 32]
  const float* w1   = (const float*)d_in[2];  // [8, 32]
  const float* b1   = (const float*)d_in[3];  // [8]
  const float* w2   = (const float*)d_in[4];  // [16, 8]
  const float* b2   = (const float*)d_in[5];  // [16]
  float* pred = (float*)d_out;

  const int E  = in_sizes[0] / 2;
  const int Nn = in_sizes[1] / 32;
  const int* src = edge;
  const int* dst = edge + E;

  // Workspace layout (floats), all slices 16B-aligned for the given sizes.
  float* ws   = (float*)d_ws;
  float* deg  = ws;                       // Nn
  float* nrm  = deg + Nn;                 // E
  float* hbuf = nrm + E;                  // Nn*16
  float* xa   = hbuf + (size_t)Nn * 16;   // Nn*16
  float* xb   = xa + (size_t)Nn * 16;     // Nn*16

  const int T = 256;
  init_deg_kernel<<<(Nn + T - 1) / T, T, 0, stream>>>(deg, Nn);
  count_deg_kernel<<<(E + T - 1) / T, T, 0, stream>>>(dst, deg, E);
  edge_norm_kernel<<<(E + T - 1) / T, T, 0, stream>>>(src, dst, deg, nrm, E);

  mlp_wmma_kernel<<<(Nn + 127) / 128, 256, 0, stream>>>(emb, w1, b1, w2, b2,
                                                        hbuf, xa, Nn);

  float* cur = xa;
  float* nxt = xb;
  for (int it = 0; it < K_STEPS; ++it) {
    node_init_kernel<<<(Nn * 4 + T - 1) / T, T, 0, stream>>>(hbuf, cur, deg, nxt, Nn);
    edge_scatter_kernel<<<(E * 4 + T - 1) / T, T, 0, stream>>>(src, dst, nrm, cur, nxt, E);
    float* tmp = cur; cur = nxt; nxt = tmp;
  }

  score_kernel<<<(E + T - 1) / T, T, 0, stream>>>(src, dst, cur, pred, E);
}